// Net_2491081031718
// MI455X (gfx1250) — compile-verified
//
#include <hip/hip_runtime.h>
#include <hip/hip_bf16.h>

// ---- model dims (fixed) ----
#define B_   8
#define T_   16
#define IND  4096
#define LAT  256
#define DM   256
#define DS_  1024
#define HD   64
#define NL   4
#define DC   4
#define DI   512
#define NH   8
#define CD   2560     // D_INNER + 2*D_STATE
#define DIP  3080     // 2*D_INNER + 2*D_STATE + NHEADS

typedef float v2f __attribute__((ext_vector_type(2)));
typedef float v8f __attribute__((ext_vector_type(8)));

__device__ __forceinline__ float dev_sigmoid(float v) { return 1.f / (1.f + expf(-v)); }
__device__ __forceinline__ float dev_silu(float v)    { return v / (1.f + expf(-v)); }
__device__ __forceinline__ float dev_softplus(float v){ return v > 20.f ? v : log1pf(expf(v)); }

// ============================================================================
// Generic skinny GEMM:  C[M,N] = act(A[M,K] @ W[N,K]^T + bias)   (TN layout)
// f32 WMMA 16x16x4. One wave per 16x16 tile, 8 waves (128 cols) per block.
// Operand lane pattern (identical for A and W since both are row-major *,K):
//   lane l: row = l&15, K pair at 2*(l>>4).
// Out-of-range rows (A: row>=M, W: n>=N) are read from clamped in-bounds
// addresses WITHOUT zeroing: garbage in A row m only affects D row m, garbage
// in W row n only affects D column n, and those D entries are never stored.
// This keeps the inner loop pure load_b64 -> v_wmma.
// ============================================================================
template<int ACT, bool RESID>
__global__ __launch_bounds__(256) void gemm_wt_kernel(
    const float* __restrict__ A, int lda,
    const float* __restrict__ W, int ldw,
    const float* __restrict__ bias,
    float* __restrict__ C, int ldc,
    int M, int N, int K)
{
    const int lane = threadIdx.x & 31;
    const int wave = threadIdx.x >> 5;
    const int n0   = (blockIdx.x * 8 + wave) * 16;
    if (n0 >= N) return;                     // wave-uniform exit, EXEC stays full
    const int row  = lane & 15;
    const int g    = lane >> 4;
    const bool aval = (row < M);
    const int  wn   = n0 + row;
    const bool wval = (wn < N);
    const float* Arow = A + (size_t)(aval ? row : 0) * lda + 2 * g;
    const float* Wrow = W + (size_t)(wval ? wn  : 0) * ldw + 2 * g;
    v8f acc = {0.f,0.f,0.f,0.f,0.f,0.f,0.f,0.f};
#pragma unroll 8
    for (int k = 0; k < K; k += 4) {
        v2f a = *(const v2f*)(Arow + k);
        v2f b = *(const v2f*)(Wrow + k);
        // D = A(16x4) x B(4x16) + C
        acc = __builtin_amdgcn_wmma_f32_16x16x4_f32(
            false, a, false, b, (short)0, acc, false, false);
    }
    // C/D layout: vgpr r -> M = r + 8*g, N = n0 + (lane&15)
    if (wval) {
        const float bv = bias ? bias[wn] : 0.f;
#pragma unroll
        for (int r = 0; r < 8; ++r) {
            const int m = r + 8 * g;
            if (m < M) {
                float v = acc[r] + bv;
                if      (ACT == 1) v = fmaxf(v, 0.f);
                else if (ACT == 2) v = dev_silu(v);
                else if (ACT == 3) v = dev_sigmoid(v);
                if (RESID) C[(size_t)m * ldc + wn] += v;
                else       C[(size_t)m * ldc + wn]  = v;
            }
        }
    }
}

// ============================================================================
// Elementwise / reduction kernels
// ============================================================================
__global__ __launch_bounds__(256) void zero_kernel(float* __restrict__ p, long n) {
    long i = (long)blockIdx.x * 256 + threadIdx.x;
    long stride = (long)gridDim.x * 256;
    for (; i < n; i += stride) p[i] = 0.f;
}

// RMSNorm over one row per block
__global__ __launch_bounds__(256) void rms_kernel(
    const float* __restrict__ in, const float* __restrict__ w,
    float* __restrict__ out, int D)
{
    __shared__ float red[256];
    const int b = blockIdx.x, tid = threadIdx.x;
    const float* xr = in + (size_t)b * D;
    float ss = 0.f;
    for (int i = tid; i < D; i += 256) { float v = xr[i]; ss += v * v; }
    red[tid] = ss; __syncthreads();
    for (int s = 128; s > 0; s >>= 1) { if (tid < s) red[tid] += red[tid + s]; __syncthreads(); }
    const float sc = rsqrtf(red[0] / (float)D + 1e-5f);
    for (int i = tid; i < D; i += 256) out[(size_t)b * D + i] = xr[i] * sc * w[i];
}

// y2 = rms(y * silu(z), w); z has row stride ldz (slice of zxbcdt), D = 512
__global__ __launch_bounds__(256) void gated_rms_kernel(
    const float* __restrict__ y, const float* __restrict__ z, int ldz,
    const float* __restrict__ w, float* __restrict__ out)
{
    __shared__ float buf[DI];
    __shared__ float red[256];
    const int b = blockIdx.x, tid = threadIdx.x;
    float ss = 0.f;
    for (int i = tid; i < DI; i += 256) {
        float zz = z[(size_t)b * ldz + i];
        float v  = y[(size_t)b * DI + i] * dev_silu(zz);
        buf[i] = v; ss += v * v;
    }
    red[tid] = ss; __syncthreads();
    for (int s = 128; s > 0; s >>= 1) { if (tid < s) red[tid] += red[tid + s]; __syncthreads(); }
    const float sc = rsqrtf(red[0] / (float)DI + 1e-5f);
    for (int i = tid; i < DI; i += 256) out[(size_t)b * DI + i] = buf[i] * sc * w[i];
}

// zlat = mu + eps*exp(0.5 lv); h := zlat; loss += -0.5*sum(1+lv-mu^2-exp(lv))
__global__ __launch_bounds__(256) void latent_kernel(
    const float* __restrict__ mu, const float* __restrict__ lv,
    const float* __restrict__ eps_t, float* __restrict__ hbuf,
    float* __restrict__ loss)
{
    __shared__ float red[256];
    const int idx = blockIdx.x * 256 + threadIdx.x;   // B_*LAT = 2048
    const int tid = threadIdx.x;
    float kl = 0.f;
    if (idx < B_ * LAT) {
        float m = mu[idx], l = lv[idx];
        hbuf[idx] = m + eps_t[idx] * expf(0.5f * l);
        kl = -0.5f * (1.f + l - m * m - expf(l));
    }
    red[tid] = kl; __syncthreads();
    for (int s = 128; s > 0; s >>= 1) { if (tid < s) red[tid] += red[tid + s]; __syncthreads(); }
    if (tid == 0) atomicAdd(loss, red[0]);
}

// loss += sum((recon - x_t)^2)
__global__ __launch_bounds__(256) void recon_loss_kernel(
    const float* __restrict__ recon, const float* __restrict__ x, int t,
    float* __restrict__ loss)
{
    __shared__ float red[256];
    const int idx = blockIdx.x * 256 + threadIdx.x;   // B_*IND = 32768
    const int tid = threadIdx.x;
    float v = 0.f;
    if (idx < B_ * IND) {
        int b = idx >> 12, i = idx & (IND - 1);
        float d = recon[idx] - x[((size_t)b * T_ + t) * IND + i];
        v = d * d;
    }
    red[tid] = v; __syncthreads();
    for (int s = 128; s > 0; s >>= 1) { if (tid < s) red[tid] += red[tid + s]; __syncthreads(); }
    if (tid == 0) atomicAdd(loss, red[0]);
}

// conv-state shift + depthwise conv + SiLU.  One thread per (b,c).
__global__ __launch_bounds__(256) void conv_kernel(
    float* __restrict__ cs_layer, const float* __restrict__ zx,
    const float* __restrict__ cw, const float* __restrict__ cb,
    float* __restrict__ xBC)
{
    const int idx = blockIdx.x * 256 + threadIdx.x;   // B_*CD = 20480
    if (idx >= B_ * CD) return;
    const int b = idx / CD, c = idx % CD;
    float* cs = cs_layer + ((size_t)b * CD + c) * DC;
    const float xin = zx[(size_t)b * DIP + DI + c];
    float s0 = cs[1], s1 = cs[2], s2 = cs[3], s3 = xin;
    cs[0] = s0; cs[1] = s1; cs[2] = s2; cs[3] = s3;
    const float* w = cw + (size_t)c * DC;
    float s = s0 * w[0] + s1 * w[1] + s2 * w[2] + s3 * w[3] + cb[c];
    xBC[(size_t)b * CD + c] = dev_silu(s);
}

// SSM state update + y reduction. One block per (b,h,p); 1024 n per block.
__global__ __launch_bounds__(256) void ssm_kernel(
    float* __restrict__ st_layer, const float* __restrict__ zx,
    const float* __restrict__ xBC, const float* __restrict__ dt_bias,
    const float* __restrict__ A_log, const float* __restrict__ Dp,
    float* __restrict__ y)
{
    __shared__ float red[256];
    const int p = blockIdx.x & (HD - 1);
    const int h = (blockIdx.x >> 6) & (NH - 1);
    const int b = blockIdx.x >> 9;
    const int tid = threadIdx.x;

    const float dtr = zx[(size_t)b * DIP + (DI + CD) + h] + dt_bias[h];
    const float dt  = dev_softplus(dtr);
    const float Av  = -expf(A_log[h]);
    const float dA  = expf(dt * Av);
    const float xv  = xBC[(size_t)b * CD + h * HD + p];
    const float* Bm = xBC + (size_t)b * CD + DI;
    const float* Cm = Bm + DS_;
    float* st = st_layer + ((((size_t)b * NH + h) * HD + p) << 10);

    float part = 0.f;
#pragma unroll
    for (int k = 0; k < 4; ++k) {
        const int n = tid + k * 256;
        float s = st[n] * dA + dt * Bm[n] * xv;
        st[n] = s;
        part += s * Cm[n];
    }
    red[tid] = part; __syncthreads();
    for (int s = 128; s > 0; s >>= 1) { if (tid < s) red[tid] += red[tid + s]; __syncthreads(); }
    if (tid == 0) y[(size_t)b * DI + h * HD + p] = red[0] + Dp[h] * xv;
}

__global__ void write_loss_kernel(const float* __restrict__ loss, float* __restrict__ out) {
    if (threadIdx.x == 0 && blockIdx.x == 0) out[0] = loss[0];
}

// ============================================================================
// Host-side launcher
// ============================================================================
static void gemm(hipStream_t st, int act, bool resid,
                 const float* A, int lda, const float* W, int ldw,
                 const float* bias, float* C, int ldc, int M, int N, int K)
{
    dim3 grid((N + 127) / 128);
    if (resid) {
        gemm_wt_kernel<2, true><<<grid, 256, 0, st>>>(A, lda, W, ldw, bias, C, ldc, M, N, K);
        return;
    }
    switch (act) {
    case 0: gemm_wt_kernel<0, false><<<grid, 256, 0, st>>>(A, lda, W, ldw, bias, C, ldc, M, N, K); break;
    case 1: gemm_wt_kernel<1, false><<<grid, 256, 0, st>>>(A, lda, W, ldw, bias, C, ldc, M, N, K); break;
    case 2: gemm_wt_kernel<2, false><<<grid, 256, 0, st>>>(A, lda, W, ldw, bias, C, ldc, M, N, K); break;
    default: gemm_wt_kernel<3, false><<<grid, 256, 0, st>>>(A, lda, W, ldw, bias, C, ldc, M, N, K); break;
    }
}

extern "C" void kernel_launch(void* const* d_in, const int* in_sizes, int n_in,
                              void* d_out, int out_size, void* d_ws, size_t ws_size,
                              hipStream_t stream)
{
    (void)in_sizes; (void)n_in; (void)out_size; (void)ws_size;
    const float* x        = (const float*)d_in[0];   // (B,T,IND)
    const float* eps      = (const float*)d_in[1];   // (T,B,LAT)
    const float* enc_w1   = (const float*)d_in[2];
    const float* enc_b1   = (const float*)d_in[3];
    const float* enc_w2   = (const float*)d_in[4];
    const float* enc_b2   = (const float*)d_in[5];
    const float* mu_w     = (const float*)d_in[6];
    const float* mu_b     = (const float*)d_in[7];
    const float* lv_w     = (const float*)d_in[8];
    const float* lv_b     = (const float*)d_in[9];
    const float* dec_w1   = (const float*)d_in[10];
    const float* dec_b1   = (const float*)d_in[11];
    const float* dec_w2   = (const float*)d_in[12];
    const float* dec_b2   = (const float*)d_in[13];
    const float* dec_w3   = (const float*)d_in[14];
    const float* dec_b3   = (const float*)d_in[15];
    const float* in_proj_w    = (const float*)d_in[16]; // (NL,DIP,DM)
    const float* conv_w       = (const float*)d_in[17]; // (NL,CD,DC)
    const float* conv_b       = (const float*)d_in[18]; // (NL,CD)
    const float* dt_bias      = (const float*)d_in[19]; // (NL,NH)
    const float* A_log        = (const float*)d_in[20]; // (NL,NH)
    const float* Dp           = (const float*)d_in[21]; // (NL,NH)
    const float* mixer_norm_w = (const float*)d_in[22]; // (NL,DI)
    const float* out_proj_w   = (const float*)d_in[23]; // (NL,DM,DI)
    const float* norm1_w      = (const float*)d_in[24]; // (NL,DM)
    const float* norm2_w      = (const float*)d_in[25]; // (NL,DM)
    const float* ff_w1        = (const float*)d_in[26]; // (NL,DM,DM)
    const float* ff_b1        = (const float*)d_in[27];
    const float* ff_w2        = (const float*)d_in[28];
    const float* ff_b2        = (const float*)d_in[29];
    const float* norm_f_w     = (const float*)d_in[30]; // (DM)
    float* out = (float*)d_out;
    float* ws  = (float*)d_ws;

    // ---- workspace layout (floats) ----
    const size_t OFF_CONV = 64;
    const size_t CONV_SZ  = (size_t)NL * B_ * CD * DC;           // 327680
    const size_t OFF_SSM  = OFF_CONV + CONV_SZ;
    const size_t SSM_SZ   = (size_t)NL * B_ * NH * HD * DS_;     // 16.8M
    size_t o = OFF_SSM + SSM_SZ;
    float* loss  = ws;
    float* convS = ws + OFF_CONV;
    float* ssmS  = ws + OFF_SSM;
    float* hbuf  = ws + o; o += B_ * DM;
    float* mu    = ws + o; o += B_ * DM;
    float* lv    = ws + o; o += B_ * DM;
    float* e1    = ws + o; o += B_ * 512;
    float* e2    = ws + o; o += B_ * 256;
    float* d1    = ws + o; o += B_ * 256;
    float* d2    = ws + o; o += B_ * 512;
    float* recon = ws + o; o += B_ * IND;
    float* ubuf  = ws + o; o += B_ * DM;
    float* zx    = ws + o; o += (size_t)B_ * DIP + 64;
    float* xBC   = ws + o; o += B_ * CD;
    float* ybuf  = ws + o; o += B_ * DI;
    float* y2    = ws + o; o += B_ * DI;
    float* ob    = ws + o; o += B_ * DM;
    float* f1    = ws + o; o += B_ * DM;
    float* hf    = ws + o; o += B_ * DM;

    // zero loss + recurrent states every call (deterministic replay)
    zero_kernel<<<4096, 256, 0, stream>>>(ws, (long)(OFF_SSM + SSM_SZ));

    for (int t = 0; t < T_; ++t) {
        const float* xt = x + (size_t)t * IND;            // row stride T_*IND
        // ---- VAE encoder ----
        gemm(stream, 1, false, xt, T_ * IND, enc_w1, IND, enc_b1, e1, 512, B_, 512, IND);
        gemm(stream, 1, false, e1, 512, enc_w2, 512, enc_b2, e2, 256, B_, 256, 512);
        gemm(stream, 0, false, e2, 256, mu_w, 256, mu_b, mu, LAT, B_, LAT, 256);
        gemm(stream, 0, false, e2, 256, lv_w, 256, lv_b, lv, LAT, B_, LAT, 256);
        latent_kernel<<<8, 256, 0, stream>>>(mu, lv, eps + (size_t)t * B_ * LAT, hbuf, loss);
        // ---- recon + loss ----
        gemm(stream, 1, false, hbuf, DM, dec_w1, LAT, dec_b1, d1, 256, B_, 256, LAT);
        gemm(stream, 1, false, d1, 256, dec_w2, 256, dec_b2, d2, 512, B_, 512, 256);
        gemm(stream, 3, false, d2, 512, dec_w3, 512, dec_b3, recon, IND, B_, IND, 512);
        recon_loss_kernel<<<128, 256, 0, stream>>>(recon, x, t, loss);
        // ---- mamba layers ----
        for (int j = 0; j < NL; ++j) {
            rms_kernel<<<B_, 256, 0, stream>>>(hbuf, norm1_w + j * DM, ubuf, DM);
            gemm(stream, 0, false, ubuf, DM, in_proj_w + (size_t)j * DIP * DM, DM,
                 nullptr, zx, DIP, B_, DIP, DM);
            conv_kernel<<<(B_ * CD + 255) / 256, 256, 0, stream>>>(
                convS + (size_t)j * B_ * CD * DC, zx,
                conv_w + (size_t)j * CD * DC, conv_b + (size_t)j * CD, xBC);
            ssm_kernel<<<B_ * NH * HD, 256, 0, stream>>>(
                ssmS + (size_t)j * B_ * NH * HD * DS_, zx, xBC,
                dt_bias + j * NH, A_log + j * NH, Dp + j * NH, ybuf);
            gated_rms_kernel<<<B_, 256, 0, stream>>>(ybuf, zx, DIP, mixer_norm_w + (size_t)j * DI, y2);
            gemm(stream, 0, false, y2, DI, out_proj_w + (size_t)j * DM * DI, DI,
                 nullptr, ob, DM, B_, DM, DI);
            rms_kernel<<<B_, 256, 0, stream>>>(ob, norm2_w + j * DM, ubuf, DM);
            gemm(stream, 2, false, ubuf, DM, ff_w1 + (size_t)j * DM * DM, DM,
                 ff_b1 + j * DM, f1, DM, B_, DM, DM);
            gemm(stream, 2, true, f1, DM, ff_w2 + (size_t)j * DM * DM, DM,
                 ff_b2 + j * DM, hbuf, DM, B_, DM, DM);   // h += silu(...)
        }
        // ---- final norm + decode to output ----
        rms_kernel<<<B_, 256, 0, stream>>>(hbuf, norm_f_w, hf, DM);
        gemm(stream, 1, false, hf, DM, dec_w1, LAT, dec_b1, d1, 256, B_, 256, LAT);
        gemm(stream, 1, false, d1, 256, dec_w2, 256, dec_b2, d2, 512, B_, 512, 256);
        gemm(stream, 3, false, d2, 512, dec_w3, 512, dec_b3,
             out + (size_t)t * IND, T_ * IND, B_, IND, 512);  // ys[b,t,:]
    }
    write_loss_kernel<<<1, 32, 0, stream>>>(loss, out + (size_t)B_ * T_ * IND);
}